// ExtractPatch_61117384622244
// MI455X (gfx1250) — compile-verified
//
#include <hip/hip_runtime.h>
#include <stdint.h>

// ---------------------------------------------------------------------------
// ExtractPatch for MI455X (gfx1250).
// One wave32 per match. Two TDM tensor_load_to_lds ops fetch the two 17x17
// f32 tiles (clamped in-bounds; zero-padding applied in registers). TDM ops
// complete in-order per wave, so we wait tensorcnt<=1, normalize patch 1
// while patch 2's DMA is in flight, then wait 0 for patch 2.
// Memory-bound: ~152 MB total traffic at 23.3 TB/s; images (64 MB) L2-resident.
// ---------------------------------------------------------------------------

namespace {
constexpr int kB = 8, kH = 1024, kW = 1024, kN = 4096;
constexpr int kR = 8, kD = 17, kP = kD * kD;   // patch = 289 elements
}

typedef unsigned int uint32;
typedef uint32 v4u __attribute__((ext_vector_type(4)));
typedef int    v8i __attribute__((ext_vector_type(8)));
typedef int    v4i __attribute__((ext_vector_type(4)));

__device__ __forceinline__ int    rfl_i(int v)    { return __builtin_amdgcn_readfirstlane(v); }
__device__ __forceinline__ uint32 rfl_u(uint32 v) { return (uint32)__builtin_amdgcn_readfirstlane((int)v); }

// Consume tile I: wait until it has landed (in-order TDM completion), compute
// mean/std over the zero-padded 17x17 patch, write normalized values coalesced.
template <int I>
__device__ __forceinline__ void process_tile(const float* tl, int mx, int my,
                                             int xc, int yc, int lane,
                                             float* __restrict__ out, size_t obase)
{
  // Tile I is ready once TENSORcnt <= (1 - I); tile 0 consumed while tile 1
  // is still streaming in.
  __builtin_amdgcn_s_wait_tensorcnt(1 - I);

  float v[10];
  float sum = 0.f, ssq = 0.f;

  #pragma unroll
  for (int u = 0; u < 10; ++u) {
    int t = lane + 32 * u;                      // flat patch index k
    float val = 0.f;
    if (t < kP) {
      int j = t / kD;                           // column offset -> x
      int r = t - j * kD;                       // row offset    -> y
      int y = my + r - kR;                      // unpadded image coords
      int x = mx + j - kR;
      if ((unsigned)y < (unsigned)kH && (unsigned)x < (unsigned)kW)
        val = tl[(y - yc) * kD + (x - xc)];
      // else: zero-padded region -> 0
    }
    v[u] = val;
    sum += val;
    ssq += val * val;
  }

  // wave32 butterfly reduction (all lanes end with totals)
  #pragma unroll
  for (int o = 16; o > 0; o >>= 1) {
    sum += __shfl_xor(sum, o, 32);
    ssq += __shfl_xor(ssq, o, 32);
  }

  const float mean = sum * (1.0f / (float)kP);
  float var = (ssq - sum * sum * (1.0f / (float)kP)) * (1.0f / (float)(kP - 1));
  var = fmaxf(var, 0.0f);
  const float inv = 1.0f / (sqrtf(var) + 1e-4f);   // (p - m) / (std + 1e-4)

  #pragma unroll
  for (int u = 0; u < 10; ++u) {
    int t = lane + 32 * u;
    if (t < kP)
      out[obase + (size_t)I * kP + t] = (v[u] - mean) * inv;  // coalesced
  }
}

__global__ __launch_bounds__(256, 4) void extract_patch_tdm(
    const float* __restrict__ img1,
    const float* __restrict__ img2,
    const int*   __restrict__ matches,
    float*       __restrict__ out)
{
  // 8 waves x 2 images x 289 floats = 18,496 B static LDS (WGP has 320 KB)
  __shared__ float tiles[8][2][kP];

  const int lane = threadIdx.x & 31;
  const int wv   = threadIdx.x >> 5;
  const int gw   = (int)blockIdx.x * 8 + wv;      // global match id [0, 32768)
  const int b    = gw >> 12;                      // batch index (N = 4096)

  // match coords are wave-uniform (one wave per match)
  const int4 m4 = *reinterpret_cast<const int4*>(matches + (size_t)gw * 4);
  int mx[2], my[2], xc[2], yc[2];
  // reference: p1 row = m[1] + k%17, col = m[0] + k//17 (padded coords)
  mx[0] = rfl_i(m4.x); my[0] = rfl_i(m4.y);
  mx[1] = rfl_i(m4.z); my[1] = rfl_i(m4.w);

  const float* imgs[2] = { img1, img2 };

  // ---- issue two TDM 2D-tile loads (17x17 f32, row stride 1024) ----------
  #pragma unroll
  for (int i = 0; i < 2; ++i) {
    // clamp tile origin so the 17x17 window is fully inside the image;
    // zero-padding is applied later with a register mask.
    xc[i] = min(max(mx[i] - kR, 0), kW - kD);
    yc[i] = min(max(my[i] - kR, 0), kH - kD);

    uint64_t ga = (uint64_t)(uintptr_t)imgs[i] +
                  4ull * ((uint64_t)b * kH * kW + (uint64_t)yc[i] * kW + (uint64_t)xc[i]);
    uint32 la = (uint32)(uintptr_t)&tiles[wv][i][0];  // LDS aperture: offset in addr[31:0]

    // D# group 0 (128b): count=1 | lds_addr | global_addr[56:0] | type=2
    v4u g0;
    g0.x = 1u;
    g0.y = rfl_u(la);
    g0.z = rfl_u((uint32)ga);
    g0.w = rfl_u((uint32)((ga >> 32) & 0x01FFFFFFull) | (2u << 30));

    // D# group 1 (256b)
    v8i g1;
    g1[0] = 0x00020000;            // workgroup_mask=0, data_size=2 (4B), no pad/iterate
    g1[1] = (int)(1024u << 16);    // atomic_barrier_addr=0 | tensor_dim0.lo16 = 1024
    g1[2] = (int)(1024u << 16);    // tensor_dim0.hi16=0    | tensor_dim1.lo16 = 1024
    g1[3] = (int)(17u << 16);      // tensor_dim1.hi16=0    | tile_dim0 = 17
    g1[4] = 17;                    // tile_dim1 = 17, tile_dim2 = 0 (2D)
    g1[5] = 1024;                  // tensor_dim0_stride.lo32 = 1024 elements
    g1[6] = 0;                     // stride0.hi16=0 | tensor_dim1_stride.lo16 (2^20 -> 0)
    g1[7] = 16;                    // tensor_dim1_stride[47:16] = 16 (stride = 2^20, unused for 2D)

    v4i g2 = (v4i)0;               // 2D tensor: groups 2/3 zero (NULL-equivalent)
    v4i g3 = (v4i)0;
    v8i g4 = (v8i)0;               // extended/tail words of the 6-arg form: zero
    __builtin_amdgcn_tensor_load_to_lds(g0, g1, g2, g3, g4, 0);
  }

  const size_t obase = (size_t)gw * (size_t)(2 * kP);

  // ---- normalize + store both patches (pipelined against 2nd DMA) --------
  process_tile<0>(&tiles[wv][0][0], mx[0], my[0], xc[0], yc[0], lane, out, obase);
  process_tile<1>(&tiles[wv][1][0], mx[1], my[1], xc[1], yc[1], lane, out, obase);
}

extern "C" void kernel_launch(void* const* d_in, const int* in_sizes, int n_in,
                              void* d_out, int out_size, void* d_ws, size_t ws_size,
                              hipStream_t stream) {
  (void)in_sizes; (void)n_in; (void)out_size; (void)d_ws; (void)ws_size;
  const float* img1    = (const float*)d_in[0];
  const float* img2    = (const float*)d_in[1];
  const int*   matches = (const int*)d_in[2];
  float*       out     = (float*)d_out;

  dim3 grid(kB * kN / 8);   // 4096 blocks x 8 waves = 32768 waves = #matches
  dim3 block(256);
  extract_patch_tdm<<<grid, block, 0, stream>>>(img1, img2, matches, out);
}